// SS2D_17626545783125
// MI455X (gfx1250) — compile-verified
//
#include <hip/hip_runtime.h>
#include <math.h>

typedef __attribute__((ext_vector_type(2))) float v2f;
typedef __attribute__((ext_vector_type(8))) float v8f;

#define B_   4
#define DM   96
#define DI   192
#define NS   16
#define RK   6
#define LTOK 4096   // 64*64
#define LP   1024   // 32*32
#define NCH  16     // chunks per sequence
#define CH   64     // chunk length (NCH*CH == LP)

__device__ __forceinline__ v8f wmma4(v2f a, v2f b, v8f c) {
  // exact-f32 WMMA: D(16x16) = A(16x4) * B(4x16) + C
  return __builtin_amdgcn_wmma_f32_16x16x4_f32(false, a, false, b, (short)0, c,
                                               false, false);
}

__device__ __forceinline__ float silu_f(float x) { return x / (1.f + __expf(-x)); }
__device__ __forceinline__ float softplus_f(float x) {
  return (x > 20.f) ? x : log1pf(__expf(x));
}

// ---------------------------------------------------------------------------
// K1: in_proj GEMM. C[16384,384] = X[16384,96] * W_in^T.  cols<192 -> xin
// (stored NCHW for the conv), cols>=192 -> z with SiLU fused.
// ---------------------------------------------------------------------------
__global__ void k_inproj(const float* __restrict__ X, const float* __restrict__ Win,
                         float* __restrict__ xin, float* __restrict__ z) {
  const int lane = threadIdx.x & 31;
  const int wv   = threadIdx.x >> 5;
  const int tile = blockIdx.x * 8 + wv;        // 1024 * 24 tiles
  const int tm = tile / 24, tn = tile % 24;
  const int hi = lane >> 4, lo = lane & 15;
  const int m = tm * 16 + lo;
  const int n = tn * 16 + lo;
  const float* xr = X   + m * DM;
  const float* wr = Win + n * DM;
  v8f acc = {0.f, 0.f, 0.f, 0.f, 0.f, 0.f, 0.f, 0.f};
  for (int k0 = 0; k0 < DM; k0 += 4) {
    const int ka = k0 + 2 * hi;
    v2f a, b;
    a.x = xr[ka]; a.y = xr[ka + 1];
    b.x = wr[ka]; b.y = wr[ka + 1];
    acc = wmma4(a, b, acc);
  }
#pragma unroll
  for (int v = 0; v < 8; ++v) {
    const int row = tm * 16 + v + 8 * hi;      // token index
    const int col = tn * 16 + lo;              // output feature
    const float val = acc[v];
    if (col < DI) {
      const int b_ = row >> 12, hw = row & 4095;
      xin[((b_ * DI + col) << 12) + hw] = val; // NCHW
    } else {
      z[row * DI + (col - DI)] = silu_f(val);  // NHWC, gate
    }
  }
}

// ---------------------------------------------------------------------------
// K2: depthwise 3x3 conv (SAME) + bias + SiLU, fused with efficient_scan
// parity scatter into xs[b][k][c][l'] (4 directions, l' = 32*32).
// ---------------------------------------------------------------------------
__global__ void k_conv_scan(const float* __restrict__ xin, const float* __restrict__ cw,
                            const float* __restrict__ cb, float* __restrict__ xs) {
  const int idx = blockIdx.x * 256 + threadIdx.x;
  if (idx >= B_ * DI * LTOK) return;
  const int w = idx & 63, h = (idx >> 6) & 63;
  const int c = (idx >> 12) % DI;
  const int b = idx / (DI * LTOK);
  const float* base = xin + ((b * DI + c) << 12);
  const float* wp = cw + c * 9;
  float acc = cb[c];
#pragma unroll
  for (int dh = -1; dh <= 1; ++dh) {
    const int h2 = h + dh;
    if ((unsigned)h2 >= 64u) continue;
#pragma unroll
    for (int dw = -1; dw <= 1; ++dw) {
      const int w2 = w + dw;
      if ((unsigned)w2 >= 64u) continue;
      acc += wp[(dh + 1) * 3 + (dw + 1)] * base[h2 * 64 + w2];
    }
  }
  const float s = silu_f(acc);
  const int hh = h >> 1, wh = w >> 1;
  const int k  = (h & 1) + ((w & 1) << 1);          // parity -> direction
  const int li = (h & 1) ? (wh * 32 + hh) : (hh * 32 + wh);
  xs[(((b * 4 + k) * DI + c) << 10) + li] = s;
}

// ---------------------------------------------------------------------------
// K3: x_proj GEMM per (b,k): dbl(38x1024) = Wp(38x192) * xs_bk(192x1024).
// M padded 38 -> 48 (zero A rows, masked writes).
// ---------------------------------------------------------------------------
__global__ void k_xproj(const float* __restrict__ xs, const float* __restrict__ xpw,
                        float* __restrict__ dbl) {
  const int lane = threadIdx.x & 31;
  const int wv   = threadIdx.x >> 5;
  const int tile = blockIdx.x * 8 + wv;        // 16 * 3 * 64 tiles
  const int bk  = tile / 192;
  const int rem = tile % 192;
  const int tmc = rem / 64, tn = rem % 64;
  const int kdir = bk & 3;
  const int hi = lane >> 4, lo = lane & 15;
  const int mc   = tmc * 16 + lo;
  const int ncol = tn * 16 + lo;
  const float* wk = xpw + kdir * (38 * DI);
  const float* bx = xs + ((size_t)(bk * DI) << 10);
  v8f acc = {0.f, 0.f, 0.f, 0.f, 0.f, 0.f, 0.f, 0.f};
  for (int k0 = 0; k0 < DI; k0 += 4) {
    const int ka = k0 + 2 * hi;
    v2f a, b;
    a.x = (mc < 38) ? wk[mc * DI + ka] : 0.f;
    a.y = (mc < 38) ? wk[mc * DI + ka + 1] : 0.f;
    b.x = bx[(ka << 10) + ncol];
    b.y = bx[((ka + 1) << 10) + ncol];
    acc = wmma4(a, b, acc);
  }
#pragma unroll
  for (int v = 0; v < 8; ++v) {
    const int c = tmc * 16 + v + 8 * hi;
    if (c < 38) dbl[((bk * 38 + c) << 10) + tn * 16 + lo] = acc[v];
  }
}

// ---------------------------------------------------------------------------
// K4: dt GEMM per (b,k): delta(192x1024) = dtW(192x6) * dbl_rows0..5(6x1024),
// K padded 6 -> 8; softplus(acc + dt_bias) fused in epilogue.
// ---------------------------------------------------------------------------
__global__ void k_dtproj(const float* __restrict__ dbl, const float* __restrict__ dtw,
                         const float* __restrict__ dtb, float* __restrict__ delta) {
  const int lane = threadIdx.x & 31;
  const int wv   = threadIdx.x >> 5;
  const int tile = blockIdx.x * 8 + wv;        // 16 * 12 * 64 tiles
  const int bk  = tile / 768;
  const int rem = tile % 768;
  const int tm = rem / 64, tn = rem % 64;
  const int kdir = bk & 3;
  const int hi = lane >> 4, lo = lane & 15;
  const int md   = tm * 16 + lo;
  const int ncol = tn * 16 + lo;
  const float* wk = dtw + kdir * (DI * RK);
  const float* bx = dbl + ((size_t)(bk * 38) << 10);
  v8f acc = {0.f, 0.f, 0.f, 0.f, 0.f, 0.f, 0.f, 0.f};
#pragma unroll
  for (int k0 = 0; k0 < 8; k0 += 4) {
    const int ka = k0 + 2 * hi;
    v2f a, b;
    a.x = (ka     < RK) ? wk[md * RK + ka]     : 0.f;
    a.y = (ka + 1 < RK) ? wk[md * RK + ka + 1] : 0.f;
    b.x = (ka     < RK) ? bx[(ka << 10) + ncol]       : 0.f;
    b.y = (ka + 1 < RK) ? bx[((ka + 1) << 10) + ncol] : 0.f;
    acc = wmma4(a, b, acc);
  }
#pragma unroll
  for (int v = 0; v < 8; ++v) {
    const int d = tm * 16 + v + 8 * hi;
    delta[((bk * DI + d) << 10) + tn * 16 + lo] =
        softplus_f(acc[v] + dtb[kdir * DI + d]);
  }
}

// ---------------------------------------------------------------------------
// K5a: chunked scan, pass A. One block per (bk, chunk); thread = channel d.
// Computes P[n] = prod(dA) and S[n] = chunk-local state (h_in = 0).
// B chunk staged in LDS (coalesced fill, broadcast reads).
// ---------------------------------------------------------------------------
__global__ void k_scan_chunk(const float* __restrict__ xs, const float* __restrict__ delta,
                             const float* __restrict__ dbl, const float* __restrict__ Alog,
                             float* __restrict__ pstate, float* __restrict__ sstate) {
  __shared__ float sB[NS * CH];
  const int bk = blockIdx.x >> 4;     // b*4 + k
  const int ch = blockIdx.x & 15;
  const int d  = threadIdx.x;         // 0..191
  const int kdir = bk & 3;
  const int l0 = ch * CH;
  for (int t = threadIdx.x; t < NS * CH; t += DI) {
    const int n = t >> 6, li = t & 63;
    sB[t] = dbl[((bk * 38 + RK + n) << 10) + l0 + li];
  }
  __syncthreads();
  const float* u  = xs    + ((size_t)(bk * DI + d) << 10) + l0;
  const float* dt = delta + ((size_t)(bk * DI + d) << 10) + l0;
  __builtin_prefetch(u + CH, 0, 1);   // warm next chunk stream (speculative)
  __builtin_prefetch(dt + CH, 0, 1);
  const int row = kdir * DI + d;
  float A[NS], P[NS], S[NS];
#pragma unroll
  for (int n = 0; n < NS; ++n) {
    A[n] = -__expf(Alog[row * NS + n]);
    P[n] = 1.f; S[n] = 0.f;
  }
  for (int l = 0; l < CH; ++l) {
    const float dl = dt[l];
    const float du = dl * u[l];
#pragma unroll
    for (int n = 0; n < NS; ++n) {
      const float dA = __expf(dl * A[n]);
      P[n] *= dA;
      S[n] = dA * S[n] + du * sB[n * CH + l];
    }
  }
  float* pp = pstate + ((size_t)blockIdx.x * DI + d) * NS;
  float* sp = sstate + ((size_t)blockIdx.x * DI + d) * NS;
#pragma unroll
  for (int n = 0; n < NS; ++n) { pp[n] = P[n]; sp[n] = S[n]; }
}

// ---------------------------------------------------------------------------
// K5b: serial prefix over the 16 chunks (cheap): h_in(c+1) = P(c)h_in(c)+S(c).
// ---------------------------------------------------------------------------
__global__ void k_scan_prefix(const float* __restrict__ pstate,
                              const float* __restrict__ sstate,
                              float* __restrict__ hin) {
  const int idx = blockIdx.x * 256 + threadIdx.x;   // 16*192 sequences
  if (idx >= 16 * DI) return;
  const int bk = idx / DI, d = idx % DI;
  float h[NS];
#pragma unroll
  for (int n = 0; n < NS; ++n) h[n] = 0.f;
  for (int ch = 0; ch < NCH; ++ch) {
    const size_t base = ((size_t)(bk * NCH + ch) * DI + d) * NS;
#pragma unroll
    for (int n = 0; n < NS; ++n) hin[base + n] = h[n];
#pragma unroll
    for (int n = 0; n < NS; ++n) h[n] = pstate[base + n] * h[n] + sstate[base + n];
  }
}

// ---------------------------------------------------------------------------
// K5c: chunked scan, pass C. Re-runs each chunk seeded with h_in, emits y.
// B and C chunks staged in LDS.
// ---------------------------------------------------------------------------
__global__ void k_scan_apply(const float* __restrict__ xs, const float* __restrict__ delta,
                             const float* __restrict__ dbl, const float* __restrict__ Alog,
                             const float* __restrict__ Dsp, const float* __restrict__ hin,
                             float* __restrict__ ys) {
  __shared__ float sB[NS * CH];
  __shared__ float sC[NS * CH];
  const int bk = blockIdx.x >> 4;
  const int ch = blockIdx.x & 15;
  const int d  = threadIdx.x;
  const int kdir = bk & 3;
  const int l0 = ch * CH;
  for (int t = threadIdx.x; t < NS * CH; t += DI) {
    const int n = t >> 6, li = t & 63;
    sB[t] = dbl[((bk * 38 + RK + n) << 10) + l0 + li];
    sC[t] = dbl[((bk * 38 + RK + NS + n) << 10) + l0 + li];
  }
  __syncthreads();
  const float* u  = xs    + ((size_t)(bk * DI + d) << 10) + l0;
  const float* dt = delta + ((size_t)(bk * DI + d) << 10) + l0;
  __builtin_prefetch(u + CH, 0, 1);
  __builtin_prefetch(dt + CH, 0, 1);
  const int row = kdir * DI + d;
  float A[NS], h[NS];
  const size_t hbase = ((size_t)blockIdx.x * DI + d) * NS;
#pragma unroll
  for (int n = 0; n < NS; ++n) {
    A[n] = -__expf(Alog[row * NS + n]);
    h[n] = hin[hbase + n];
  }
  const float Dv = Dsp[row];
  float* yo = ys + ((size_t)(bk * DI + d) << 10) + l0;
  for (int l = 0; l < CH; ++l) {
    const float dl = dt[l], ul = u[l];
    const float du = dl * ul;
    float y = 0.f;
#pragma unroll
    for (int n = 0; n < NS; ++n) {
      const float dA = __expf(dl * A[n]);
      h[n] = dA * h[n] + du * sB[n * CH + l];
      y += h[n] * sC[n * CH + l];
    }
    yo[l] = y + ul * Dv;
  }
}

// ---------------------------------------------------------------------------
// K6: efficient_merge gather + LayerNorm(192) + gate with z. One block/token.
// ---------------------------------------------------------------------------
__global__ void k_merge_ln_gate(const float* __restrict__ ys, const float* __restrict__ z,
                                const float* __restrict__ g, const float* __restrict__ bt,
                                float* __restrict__ y) {
  __shared__ float ss[6], sq[6];
  __shared__ float smu, srs;
  const int t = blockIdx.x;           // b*4096 + h*64 + w
  const int c = threadIdx.x;          // 0..191
  const int b = t >> 12, h = (t >> 6) & 63, w = t & 63;
  const int hh = h >> 1, wh = w >> 1;
  const int k  = (h & 1) + ((w & 1) << 1);
  const int li = (h & 1) ? (wh * 32 + hh) : (hh * 32 + wh);
  const float val = ys[(((b * 4 + k) * DI + c) << 10) + li];
  float v1 = val, v2 = val * val;
#pragma unroll
  for (int off = 16; off; off >>= 1) {
    v1 += __shfl_down(v1, off, 32);
    v2 += __shfl_down(v2, off, 32);
  }
  const int lane = threadIdx.x & 31, wv = threadIdx.x >> 5;
  if (lane == 0) { ss[wv] = v1; sq[wv] = v2; }
  __syncthreads();
  if (threadIdx.x == 0) {
    float s = 0.f, q = 0.f;
    for (int i = 0; i < 6; ++i) { s += ss[i]; q += sq[i]; }
    const float mu = s * (1.f / DI);
    smu = mu;
    srs = rsqrtf(q * (1.f / DI) - mu * mu + 1e-5f);
  }
  __syncthreads();
  const float yn = (val - smu) * srs * g[c] + bt[c];
  y[t * DI + c] = yn * z[t * DI + c];
}

// ---------------------------------------------------------------------------
// K7: out_proj GEMM. out[16384,96] = y[16384,192] * W_out^T.
// ---------------------------------------------------------------------------
__global__ void k_outproj(const float* __restrict__ y, const float* __restrict__ Wout,
                          float* __restrict__ out) {
  const int lane = threadIdx.x & 31;
  const int wv   = threadIdx.x >> 5;
  const int tile = blockIdx.x * 8 + wv;        // 1024 * 6 tiles
  const int tm = tile / 6, tn = tile % 6;
  const int hi = lane >> 4, lo = lane & 15;
  const int m = tm * 16 + lo, n = tn * 16 + lo;
  const float* yr = y + m * DI;
  const float* wr = Wout + n * DI;
  v8f acc = {0.f, 0.f, 0.f, 0.f, 0.f, 0.f, 0.f, 0.f};
  for (int k0 = 0; k0 < DI; k0 += 4) {
    const int ka = k0 + 2 * hi;
    v2f a, b;
    a.x = yr[ka]; a.y = yr[ka + 1];
    b.x = wr[ka]; b.y = wr[ka + 1];
    acc = wmma4(a, b, acc);
  }
#pragma unroll
  for (int v = 0; v < 8; ++v)
    out[(tm * 16 + v + 8 * hi) * DM + tn * 16 + lo] = acc[v];
}

// ---------------------------------------------------------------------------
extern "C" void kernel_launch(void* const* d_in, const int* in_sizes, int n_in,
                              void* d_out, int out_size, void* d_ws, size_t ws_size,
                              hipStream_t stream) {
  const float* x    = (const float*)d_in[0];
  const float* Win  = (const float*)d_in[1];
  const float* cw   = (const float*)d_in[2];
  const float* cb   = (const float*)d_in[3];
  const float* xpw  = (const float*)d_in[4];
  const float* dtw  = (const float*)d_in[5];
  const float* dtb  = (const float*)d_in[6];
  const float* Alog = (const float*)d_in[7];
  const float* Dsp  = (const float*)d_in[8];
  const float* lng  = (const float*)d_in[9];
  const float* lnb  = (const float*)d_in[10];
  const float* Wout = (const float*)d_in[11];
  float* out = (float*)d_out;

  float* ws     = (float*)d_ws;
  float* xin    = ws;                  // B*192*64*64      = 3,145,728
  float* z      = xin    + 3145728;    // B*4096*192       = 3,145,728
  float* xsb    = z      + 3145728;    // B*4*192*1024     = 3,145,728
  float* dbl    = xsb    + 3145728;    // B*4*38*1024      =   622,592
  float* delta  = dbl    + 622592;     // B*4*192*1024     = 3,145,728
  float* ysb    = delta  + 3145728;    // B*4*192*1024     = 3,145,728
  float* ybuf   = ysb    + 3145728;    // B*4096*192       = 3,145,728
  float* pstate = ybuf   + 3145728;    // 16*16*192*16     =   786,432
  float* sstate = pstate + 786432;     //                  =   786,432
  float* hin    = sstate + 786432;     //                  =   786,432

  const dim3 blk(256);
  k_inproj       <<<3072,  blk, 0, stream>>>(x, Win, xin, z);
  k_conv_scan    <<<(B_ * DI * LTOK + 255) / 256, blk, 0, stream>>>(xin, cw, cb, xsb);
  k_xproj        <<<384,   blk, 0, stream>>>(xsb, xpw, dbl);
  k_dtproj       <<<1536,  blk, 0, stream>>>(dbl, dtw, dtb, delta);
  k_scan_chunk   <<<256,   dim3(DI), 0, stream>>>(xsb, delta, dbl, Alog, pstate, sstate);
  k_scan_prefix  <<<12,    blk, 0, stream>>>(pstate, sstate, hin);
  k_scan_apply   <<<256,   dim3(DI), 0, stream>>>(xsb, delta, dbl, Alog, Dsp, hin, ysb);
  k_merge_ln_gate<<<16384, dim3(DI), 0, stream>>>(ysb, z, lng, lnb, ybuf);
  k_outproj      <<<768,   blk, 0, stream>>>(ybuf, Wout, out);
}